// NSGT_SL_55396488184188
// MI455X (gfx1250) — compile-verified
//
#include <hip/hip_runtime.h>
#include <math.h>

// ============================================================================
// NSGT forward transform for MI455X (gfx1250, wave32, WMMA).
//
// Four-step FFT formulation so ~all FLOPs run on V_WMMA_F32_16X16X4_F32
// (fp32 in / fp32 accumulate -- required precision for a 264600-point FFT):
//
//   T = N1*N2 (504*525), L = maxLg = L1*L2 (84*96), derived on host.
//   pack:  xp   = padded row-major A-layout of x (zeros in pads)
//          Gp/ip = padded A-layout of G/idx
//   1. Y   = xp * W_N2            (real-A GEMM, 8 WMMA/k-step)
//   2. Y  *= exp(-2pi i m n / T)
//   3. ft  = W_N1 * Y             (complex GEMM, 16 WMMA/k-step) -> compact
//   4. Y2  = (ft[ip]*Gp) * W_L2   (gather fused into A-tile load)
//   5. Y2 *= exp(+2pi i m n / L)
//   6. out = W_L1 * Y2            (100/L folded into W_L1; re/im interleaved
//                                  stored straight into d_out)
//
// All padded dims are multiples of 32; k-loops have NO conditionals:
//   A tiles: row-major padded  -> one b64 (v2f) load per lane per operand
//   B tiles: row-pair layout   -> one b64 (v2f) load per lane per operand
// Each wave owns a 32x32 output tile (2x2 WMMA subtiles, 8 v8f accumulators).
// ============================================================================

typedef __attribute__((ext_vector_type(2))) float v2f;
typedef __attribute__((ext_vector_type(8))) float v8f;

#define TWO_PI 6.28318530717958647692f

__device__ __forceinline__ v8f wmma4(v2f a, v2f b, v8f c) {
  return __builtin_amdgcn_wmma_f32_16x16x4_f32(false, a, false, b, (short)0, c,
                                               false, false);
}

// complex MAC: (cr,ci) += (ar+i*ai) x (br+i*bi)
__device__ __forceinline__ void cmac(v8f& cr, v8f& ci, v2f ar, v2f ai,
                                     v2f br, v2f bi) {
  v2f an;
  an.x = -ai.x;
  an.y = -ai.y;
  cr = wmma4(ar, br, cr);
  cr = wmma4(an, bi, cr);
  ci = wmma4(ar, bi, ci);
  ci = wmma4(ai, br, ci);
}

// store one 16x16 f32 C/D tile into row-pair layout (unconditional)
__device__ __forceinline__ void store_pair(float* __restrict__ base,
                                           const v8f& c, int m0, int col,
                                           int Np, int half) {
#pragma unroll
  for (int v = 0; v < 8; ++v) {
    const int r = m0 + v + 8 * half;
    base[((size_t)(r >> 1) * Np + col) * 2 + (r & 1)] = c[v];
  }
}

// ---------------------------------------------------------------------------
// Packing / fill kernels
// ---------------------------------------------------------------------------

// xp[b][m*Kp + k] = (m<N1 && k<N2) ? x[b*T + m + N1*k] : 0
__global__ void k_pack_x(const float* __restrict__ x, float* __restrict__ xp,
                         int N1, int N2, int Mp, int Kp, int T,
                         long long total) {
  long long e = (long long)blockIdx.x * blockDim.x + threadIdx.x;
  if (e >= total) return;
  const long long per = (long long)Mp * Kp;
  const int b = (int)(e / per);
  const int q = (int)(e - (long long)b * per);
  const int m = q / Kp, k = q - (q / Kp) * Kp;
  xp[e] = (m < N1 && k < N2) ? x[(size_t)b * T + m + (size_t)N1 * k] : 0.f;
}

// padded A-layout copies of G and idx:
//   Gp[bin][m*Kp + k] = (m<L1 && k<L2) ? G[bin*L + m + L1*k] : 0   (idx same)
__global__ void k_pack_gidx(const float* __restrict__ G,
                            const int* __restrict__ idx,
                            float* __restrict__ Gp, int* __restrict__ ip,
                            int L1, int L2, int Mp, int Kp, int L,
                            long long total) {
  long long e = (long long)blockIdx.x * blockDim.x + threadIdx.x;
  if (e >= total) return;
  const long long per = (long long)Mp * Kp;
  const int bin = (int)(e / per);
  const int q = (int)(e - (long long)bin * per);
  const int m = q / Kp, k = q - (q / Kp) * Kp;
  const bool valid = (m < L1) && (k < L2);
  const size_t src = (size_t)bin * L + m + (size_t)L1 * k;
  Gp[e] = valid ? G[src] : 0.f;
  ip[e] = valid ? idx[src] : 0;
}

// plain row-major padded DFT matrix (used as A operand):
//   W[r*P + c] = (r<N && c<N) ? scale*cis(dir*2pi*(r*c mod N)/N) : 0
__global__ void k_fill_plain(float* __restrict__ wre, float* __restrict__ wim,
                             int P, int N, float dir, float scale) {
  long long i = (long long)blockIdx.x * blockDim.x + threadIdx.x;
  if (i >= (long long)P * P) return;
  const int r = (int)(i / P), c = (int)(i % P);
  float re = 0.f, im = 0.f;
  if (r < N && c < N) {
    const long long t = (long long)r * c % N;
    const float ang = dir * TWO_PI * (float)t / (float)N;
    float s, co;
    __sincosf(ang, &s, &co);
    re = scale * co;
    im = scale * s;
  }
  wre[i] = re;
  wim[i] = im;
}

// row-pair layout padded DFT matrix (used as B operand):
//   flat e == ((k>>1)*P + n)*2 + (k&1)
__global__ void k_fill_pair(float* __restrict__ wre, float* __restrict__ wim,
                            int P, int N, float dir, float scale) {
  long long e = (long long)blockIdx.x * blockDim.x + threadIdx.x;
  if (e >= (long long)P * P) return;
  const int kpair = (int)(e / (2 * P));
  const int rem = (int)(e - (long long)kpair * (2 * P));
  const int n = rem >> 1;
  const int k = 2 * kpair + (rem & 1);
  float re = 0.f, im = 0.f;
  if (k < N && n < N) {
    const long long t = (long long)k * n % N;
    const float ang = dir * TWO_PI * (float)t / (float)N;
    float s, co;
    __sincosf(ang, &s, &co);
    re = scale * co;
    im = scale * s;
  }
  wre[e] = re;
  wim[e] = im;
}

// in-place twiddle on a batched row-pair buffer:
//   element (m,n) *= cis(dir*2pi*(m*n mod Ntot)/Ntot)
__global__ void k_twiddle_pair(float* __restrict__ yre, float* __restrict__ yim,
                               long long total, long long perBatch, int Np,
                               int Ntot, float dir) {
  long long e = (long long)blockIdx.x * blockDim.x + threadIdx.x;
  if (e >= total) return;
  const long long q = e % perBatch;
  const int kpair = (int)(q / (2 * Np));
  const int rem = (int)(q - (long long)kpair * (2 * Np));
  const int n = rem >> 1;
  const int m = 2 * kpair + (rem & 1);
  const long long t = (long long)m * n % Ntot;
  const float ang = dir * TWO_PI * (float)t / (float)Ntot;
  float s, co;
  __sincosf(ang, &s, &co);
  const float xr = yre[e], xi = yim[e];
  yre[e] = xr * co - xi * s;
  yim[e] = xr * s + xi * co;
}

// ---------------------------------------------------------------------------
// GEMM kernels: one 32x32 output tile per wave, no conditionals in k-loop.
// Lane mapping (wave32): half=lane>>4, l16=lane&15
//   A 16x4 : lane half owns K={k0+2h, k0+2h+1}  -> v2f from row-major A
//   B 4x16 : same K split, N=l16               -> v2f from row-pair B
//   C/D    : VGPR v -> row v+8h, col l16
// ---------------------------------------------------------------------------

// Stage 1 (fwd FFT step A): real A. D -> row-pair Y (batched).
__global__ void k_gemm_rA(const float* __restrict__ A,   // xp [b][Mp x Kp]
                          const float* __restrict__ Bre, // pair [Kp x Np]
                          const float* __restrict__ Bim,
                          float* __restrict__ Dre,       // pair [b][Mp x Np]
                          float* __restrict__ Dim_, int Mp, int Kp, int Np) {
  const int lane = threadIdx.x & 31;
  const int half = lane >> 4;
  const int l16 = lane & 15;
  const int wv = threadIdx.x >> 5;
  const int b = blockIdx.z;
  const int m0 = blockIdx.y * 32;
  const int n0 = (blockIdx.x * (blockDim.x >> 5) + wv) * 32;
  if (n0 >= Np) return;  // wave-uniform: EXEC stays all-ones elsewhere
  const size_t aBat = (size_t)Mp * Kp;
  const v2f* a0 = (const v2f*)(A + (size_t)b * aBat + (size_t)(m0 + l16) * Kp);
  const v2f* a1 = (const v2f*)(A + (size_t)b * aBat +
                               (size_t)(m0 + 16 + l16) * Kp);
  const v2f* br = (const v2f*)Bre;
  const v2f* bi = (const v2f*)Bim;
  const int col = n0 + l16;
  v8f cr00 = {}, cr01 = {}, cr10 = {}, cr11 = {};
  v8f ci00 = {}, ci01 = {}, ci10 = {}, ci11 = {};
  for (int k0 = 0; k0 < Kp; k0 += 4) {
    const int kh = (k0 >> 1) + half;
    const v2f aA = a0[kh];
    const v2f aB = a1[kh];
    const size_t bo = (size_t)kh * Np + col;
    const v2f b0r = br[bo], b1r = br[bo + 16];
    const v2f b0i = bi[bo], b1i = bi[bo + 16];
    cr00 = wmma4(aA, b0r, cr00);
    cr01 = wmma4(aA, b1r, cr01);
    cr10 = wmma4(aB, b0r, cr10);
    cr11 = wmma4(aB, b1r, cr11);
    ci00 = wmma4(aA, b0i, ci00);
    ci01 = wmma4(aA, b1i, ci01);
    ci10 = wmma4(aB, b0i, ci10);
    ci11 = wmma4(aB, b1i, ci11);
  }
  const size_t dBat = (size_t)Mp * Np;
  float* dr = Dre + (size_t)b * dBat;
  float* di = Dim_ + (size_t)b * dBat;
  store_pair(dr, cr00, m0, col, Np, half);
  store_pair(dr, cr01, m0, col + 16, Np, half);
  store_pair(dr, cr10, m0 + 16, col, Np, half);
  store_pair(dr, cr11, m0 + 16, col + 16, Np, half);
  store_pair(di, ci00, m0, col, Np, half);
  store_pair(di, ci01, m0, col + 16, Np, half);
  store_pair(di, ci10, m0 + 16, col, Np, half);
  store_pair(di, ci11, m0 + 16, col + 16, Np, half);
}

// Stage 3 (fwd FFT step B): complex shared A (W_N1 plain padded),
// batched row-pair B (Y), compact guarded planar D (ft, Mlog x Nlog).
__global__ void k_gemm_ccA_compact(
    const float* __restrict__ Are, const float* __restrict__ Aim,
    const float* __restrict__ Bre, const float* __restrict__ Bim,
    float* __restrict__ Dre, float* __restrict__ Dim_, int Kp, int Np,
    int Mlog, int Nlog) {
  const int lane = threadIdx.x & 31;
  const int half = lane >> 4;
  const int l16 = lane & 15;
  const int wv = threadIdx.x >> 5;
  const int b = blockIdx.z;
  const int m0 = blockIdx.y * 32;
  const int n0 = (blockIdx.x * (blockDim.x >> 5) + wv) * 32;
  if (n0 >= Np) return;
  const v2f* ar0 = (const v2f*)(Are + (size_t)(m0 + l16) * Kp);
  const v2f* ar1 = (const v2f*)(Are + (size_t)(m0 + 16 + l16) * Kp);
  const v2f* ai0 = (const v2f*)(Aim + (size_t)(m0 + l16) * Kp);
  const v2f* ai1 = (const v2f*)(Aim + (size_t)(m0 + 16 + l16) * Kp);
  const size_t bBat = (size_t)Kp * Np;
  const v2f* br = (const v2f*)(Bre + (size_t)b * bBat);
  const v2f* bi = (const v2f*)(Bim + (size_t)b * bBat);
  const int col = n0 + l16;
  v8f cr00 = {}, cr01 = {}, cr10 = {}, cr11 = {};
  v8f ci00 = {}, ci01 = {}, ci10 = {}, ci11 = {};
  for (int k0 = 0; k0 < Kp; k0 += 4) {
    const int kh = (k0 >> 1) + half;
    const v2f aAr = ar0[kh], aAi = ai0[kh];
    const v2f aBr = ar1[kh], aBi = ai1[kh];
    const size_t bo = (size_t)kh * Np + col;
    const v2f b0r = br[bo], b1r = br[bo + 16];
    const v2f b0i = bi[bo], b1i = bi[bo + 16];
    cmac(cr00, ci00, aAr, aAi, b0r, b0i);
    cmac(cr01, ci01, aAr, aAi, b1r, b1i);
    cmac(cr10, ci10, aBr, aBi, b0r, b0i);
    cmac(cr11, ci11, aBr, aBi, b1r, b1i);
  }
  const size_t dBat = (size_t)Mlog * Nlog;
  const v8f* accR[4] = {&cr00, &cr01, &cr10, &cr11};
  const v8f* accI[4] = {&ci00, &ci01, &ci10, &ci11};
#pragma unroll
  for (int s = 0; s < 4; ++s) {
    const int mo = m0 + (s >> 1) * 16;
    const int co = col + (s & 1) * 16;
    if (co < Nlog) {
#pragma unroll
      for (int v = 0; v < 8; ++v) {
        const int r = mo + v + 8 * half;
        if (r < Mlog) {
          const size_t off = (size_t)b * dBat + (size_t)r * Nlog + co;
          Dre[off] = (*accR[s])[v];
          Dim_[off] = (*accI[s])[v];
        }
      }
    }
  }
}

// Stage 4 (inv FFT step A): NSGT gather fused into A-tile load.
//   A(m,k) = ft[sig][ ip[bin][m*Kp+k] ] * Gp[bin][m*Kp+k]   (pads: G=0)
//   B = W_L2 row-pair; D -> row-pair Y2 (batched by r = sig*nbins+bin).
__global__ void k_gemm_gather(const float* __restrict__ ftre,
                              const float* __restrict__ ftim,
                              const float* __restrict__ Gp,
                              const int* __restrict__ ip,
                              const float* __restrict__ Bre,
                              const float* __restrict__ Bim,
                              float* __restrict__ Dre, float* __restrict__ Dim_,
                              int Kp, int Np, int Mp, int nbins, int T) {
  const int lane = threadIdx.x & 31;
  const int half = lane >> 4;
  const int l16 = lane & 15;
  const int wv = threadIdx.x >> 5;
  const int r = blockIdx.z;
  const int sig = r / nbins;
  const int bin = r - sig * nbins;
  const int m0 = blockIdx.y * 32;
  const int n0 = (blockIdx.x * (blockDim.x >> 5) + wv) * 32;
  if (n0 >= Np) return;
  const size_t aBat = (size_t)Mp * Kp;
  const v2f* g0 = (const v2f*)(Gp + (size_t)bin * aBat + (size_t)(m0 + l16) * Kp);
  const v2f* g1 =
      (const v2f*)(Gp + (size_t)bin * aBat + (size_t)(m0 + 16 + l16) * Kp);
  const int* i0 = ip + (size_t)bin * aBat + (size_t)(m0 + l16) * Kp;
  const int* i1 = ip + (size_t)bin * aBat + (size_t)(m0 + 16 + l16) * Kp;
  const float* fre = ftre + (size_t)sig * T;
  const float* fim = ftim + (size_t)sig * T;
  const v2f* br = (const v2f*)Bre;
  const v2f* bi = (const v2f*)Bim;
  const int col = n0 + l16;
  v8f cr00 = {}, cr01 = {}, cr10 = {}, cr11 = {};
  v8f ci00 = {}, ci01 = {}, ci10 = {}, ci11 = {};
  for (int k0 = 0; k0 < Kp; k0 += 4) {
    const int kh = (k0 >> 1) + half;
    const int ka = 2 * kh;
    const v2f gA = g0[kh], gB = g1[kh];
    const int tA0 = i0[ka], tA1 = i0[ka + 1];
    const int tB0 = i1[ka], tB1 = i1[ka + 1];
    v2f aAr, aAi, aBr, aBi;
    aAr.x = fre[tA0] * gA.x;
    aAr.y = fre[tA1] * gA.y;
    aAi.x = fim[tA0] * gA.x;
    aAi.y = fim[tA1] * gA.y;
    aBr.x = fre[tB0] * gB.x;
    aBr.y = fre[tB1] * gB.y;
    aBi.x = fim[tB0] * gB.x;
    aBi.y = fim[tB1] * gB.y;
    const size_t bo = (size_t)kh * Np + col;
    const v2f b0r = br[bo], b1r = br[bo + 16];
    const v2f b0i = bi[bo], b1i = bi[bo + 16];
    cmac(cr00, ci00, aAr, aAi, b0r, b0i);
    cmac(cr01, ci01, aAr, aAi, b1r, b1i);
    cmac(cr10, ci10, aBr, aBi, b0r, b0i);
    cmac(cr11, ci11, aBr, aBi, b1r, b1i);
  }
  const size_t dBat = (size_t)Mp * Np;
  float* dr = Dre + (size_t)r * dBat;
  float* di = Dim_ + (size_t)r * dBat;
  store_pair(dr, cr00, m0, col, Np, half);
  store_pair(dr, cr01, m0, col + 16, Np, half);
  store_pair(dr, cr10, m0 + 16, col, Np, half);
  store_pair(dr, cr11, m0 + 16, col + 16, Np, half);
  store_pair(di, ci00, m0, col, Np, half);
  store_pair(di, ci01, m0, col + 16, Np, half);
  store_pair(di, ci10, m0 + 16, col, Np, half);
  store_pair(di, ci11, m0 + 16, col + 16, Np, half);
}

// Stage 6 (inv FFT step B): complex shared A (W_L1, 100/L folded in),
// batched row-pair B (Y2), guarded interleaved store into d_out:
//   out[(r*Llog + m*Nlog + n)*2 + {0,1}]
__global__ void k_gemm_ccA_out(const float* __restrict__ Are,
                               const float* __restrict__ Aim,
                               const float* __restrict__ Bre,
                               const float* __restrict__ Bim,
                               float* __restrict__ out, int Kp, int Np,
                               int Mlog, int Nlog, long long Llog) {
  const int lane = threadIdx.x & 31;
  const int half = lane >> 4;
  const int l16 = lane & 15;
  const int wv = threadIdx.x >> 5;
  const int b = blockIdx.z;
  const int m0 = blockIdx.y * 32;
  const int n0 = (blockIdx.x * (blockDim.x >> 5) + wv) * 32;
  if (n0 >= Np) return;
  const v2f* ar0 = (const v2f*)(Are + (size_t)(m0 + l16) * Kp);
  const v2f* ar1 = (const v2f*)(Are + (size_t)(m0 + 16 + l16) * Kp);
  const v2f* ai0 = (const v2f*)(Aim + (size_t)(m0 + l16) * Kp);
  const v2f* ai1 = (const v2f*)(Aim + (size_t)(m0 + 16 + l16) * Kp);
  const size_t bBat = (size_t)Kp * Np;
  const v2f* br = (const v2f*)(Bre + (size_t)b * bBat);
  const v2f* bi = (const v2f*)(Bim + (size_t)b * bBat);
  const int col = n0 + l16;
  v8f cr00 = {}, cr01 = {}, cr10 = {}, cr11 = {};
  v8f ci00 = {}, ci01 = {}, ci10 = {}, ci11 = {};
  for (int k0 = 0; k0 < Kp; k0 += 4) {
    const int kh = (k0 >> 1) + half;
    const v2f aAr = ar0[kh], aAi = ai0[kh];
    const v2f aBr = ar1[kh], aBi = ai1[kh];
    const size_t bo = (size_t)kh * Np + col;
    const v2f b0r = br[bo], b1r = br[bo + 16];
    const v2f b0i = bi[bo], b1i = bi[bo + 16];
    cmac(cr00, ci00, aAr, aAi, b0r, b0i);
    cmac(cr01, ci01, aAr, aAi, b1r, b1i);
    cmac(cr10, ci10, aBr, aBi, b0r, b0i);
    cmac(cr11, ci11, aBr, aBi, b1r, b1i);
  }
  const v8f* accR[4] = {&cr00, &cr01, &cr10, &cr11};
  const v8f* accI[4] = {&ci00, &ci01, &ci10, &ci11};
#pragma unroll
  for (int s = 0; s < 4; ++s) {
    const int mo = m0 + (s >> 1) * 16;
    const int co = col + (s & 1) * 16;
    if (co < Nlog) {
#pragma unroll
      for (int v = 0; v < 8; ++v) {
        const int r = mo + v + 8 * half;
        if (r < Mlog) {
          const size_t off =
              ((size_t)b * Llog + (size_t)r * Nlog + co) * 2;
          out[off] = (*accR[s])[v];
          out[off + 1] = (*accI[s])[v];
        }
      }
    }
  }
}

// ---------------------------------------------------------------------------
// Host side
// ---------------------------------------------------------------------------
static void factor2(int n, int* f1, int* f2) {
  int r = (int)floor(sqrt((double)n));
  for (int d = r; d >= 1; --d) {
    if (n % d == 0) { *f1 = d; *f2 = n / d; return; }
  }
  *f1 = 1; *f2 = n;
}
static inline int pad32(int x) { return (x + 31) & ~31; }

extern "C" void kernel_launch(void* const* d_in, const int* in_sizes, int n_in,
                              void* d_out, int out_size, void* d_ws,
                              size_t ws_size, hipStream_t stream) {
  (void)n_in; (void)out_size; (void)ws_size;
  const float* x = (const float*)d_in[0];
  const float* G = (const float*)d_in[1];
  const int* idx = (const int*)d_in[2];
  float* out = (float*)d_out;

  const int SC = 32;                   // S*C signals
  const int T = in_sizes[0] / SC;      // 264600
  const int NBINS = 114;               // fixed by reference constants
  const int L = in_sizes[1] / NBINS;   // maxLg
  const int R = SC * NBINS;            // batched IFFT rows

  int N1, N2; factor2(T, &N1, &N2);    // 504 x 525
  int L1, L2; factor2(L, &L1, &L2);    // 84 x 96
  const int N1p = pad32(N1), N2p = pad32(N2);  // 512, 544
  const int L1p = pad32(L1), L2p = pad32(L2);  // 96, 96

  // --- carve workspace (floats, 256B-aligned regions)
  float* w = (float*)d_ws;
  size_t off = 0;
  auto carve = [&](size_t n) {
    float* p = w + off;
    off += (n + 63) & ~(size_t)63;
    return p;
  };
  float* wn2re = carve((size_t)N2p * N2p);   // pair (B of stage 1)
  float* wn2im = carve((size_t)N2p * N2p);
  float* wn1re = carve((size_t)N1p * N1p);   // plain (A of stage 3)
  float* wn1im = carve((size_t)N1p * N1p);
  float* wl2re = carve((size_t)L2p * L2p);   // pair (B of stage 4)
  float* wl2im = carve((size_t)L2p * L2p);
  float* wl1re = carve((size_t)L1p * L1p);   // plain (A of stage 6)
  float* wl1im = carve((size_t)L1p * L1p);
  float* xp    = carve((size_t)SC * N1p * N2p);
  float* yre   = carve((size_t)SC * N1p * N2p);  // pair
  float* yim   = carve((size_t)SC * N1p * N2p);
  float* fre   = carve((size_t)SC * T);          // compact
  float* fim   = carve((size_t)SC * T);
  float* Gp    = carve((size_t)NBINS * L1p * L2p);
  int*   ip    = (int*)carve((size_t)NBINS * L1p * L2p);
  float* y2re  = carve((size_t)R * L1p * L2p);   // pair
  float* y2im  = carve((size_t)R * L1p * L2p);

  const int TH = 256;
  const int WPB = 4;  // waves (32x32 tiles) per block

  // --- DFT matrices
  {
    long long t;
    t = (long long)N2p * N2p;
    k_fill_pair<<<dim3((unsigned)((t + TH - 1) / TH)), TH, 0, stream>>>(
        wn2re, wn2im, N2p, N2, -1.f, 1.f);
    t = (long long)N1p * N1p;
    k_fill_plain<<<dim3((unsigned)((t + TH - 1) / TH)), TH, 0, stream>>>(
        wn1re, wn1im, N1p, N1, -1.f, 1.f);
    t = (long long)L2p * L2p;
    k_fill_pair<<<dim3((unsigned)((t + TH - 1) / TH)), TH, 0, stream>>>(
        wl2re, wl2im, L2p, L2, 1.f, 1.f);
    t = (long long)L1p * L1p;
    k_fill_plain<<<dim3((unsigned)((t + TH - 1) / TH)), TH, 0, stream>>>(
        wl1re, wl1im, L1p, L1, 1.f, 100.f / (float)L);
  }
  // --- pack x and G/idx into padded A-layouts
  {
    long long tot = (long long)SC * N1p * N2p;
    k_pack_x<<<dim3((unsigned)((tot + TH - 1) / TH)), TH, 0, stream>>>(
        x, xp, N1, N2, N1p, N2p, T, tot);
    tot = (long long)NBINS * L1p * L2p;
    k_pack_gidx<<<dim3((unsigned)((tot + TH - 1) / TH)), TH, 0, stream>>>(
        G, idx, Gp, ip, L1, L2, L1p, L2p, L, tot);
  }
  // 1. Y = xp * W_N2
  {
    dim3 g((N2p / 32 + WPB - 1) / WPB, N1p / 32, SC);
    k_gemm_rA<<<g, 32 * WPB, 0, stream>>>(xp, wn2re, wn2im, yre, yim, N1p, N2p,
                                          N2p);
  }
  // 2. forward twiddle on Y (pair layout)
  {
    const long long per = (long long)N1p * N2p, tot = per * SC;
    k_twiddle_pair<<<dim3((unsigned)((tot + TH - 1) / TH)), TH, 0, stream>>>(
        yre, yim, tot, per, N2p, T, -1.f);
  }
  // 3. ft = W_N1 * Y  -> compact [N1 x N2]
  {
    dim3 g((N2p / 32 + WPB - 1) / WPB, N1p / 32, SC);
    k_gemm_ccA_compact<<<g, 32 * WPB, 0, stream>>>(wn1re, wn1im, yre, yim, fre,
                                                   fim, N1p, N2p, N1, N2);
  }
  // 4. Y2 = (ft[idx]*G) * W_L2   (gather fused)
  {
    dim3 g((L2p / 32 + WPB - 1) / WPB, L1p / 32, R);
    k_gemm_gather<<<g, 32 * WPB, 0, stream>>>(fre, fim, Gp, ip, wl2re, wl2im,
                                              y2re, y2im, L2p, L2p, L1p, NBINS,
                                              T);
  }
  // 5. inverse twiddle on Y2 (pair layout)
  {
    const long long per = (long long)L1p * L2p, tot = per * R;
    k_twiddle_pair<<<dim3((unsigned)((tot + TH - 1) / TH)), TH, 0, stream>>>(
        y2re, y2im, tot, per, L2p, L, 1.f);
  }
  // 6. out = W_L1 * Y2 -> interleaved re/im, scale baked into W_L1
  {
    dim3 g((L2p / 32 + WPB - 1) / WPB, L1p / 32, R);
    k_gemm_ccA_out<<<g, 32 * WPB, 0, stream>>>(wl1re, wl1im, y2re, y2im, out,
                                               L1p, L2p, L1, L2, (long long)L);
  }
}